// SSIMLoss_82660940579577
// MI455X (gfx1250) — compile-verified
//
#include <hip/hip_runtime.h>
#include <cmath>
#include <cstdint>

typedef float v2f __attribute__((ext_vector_type(2)));
typedef float v4f __attribute__((ext_vector_type(4)));
typedef float v8f __attribute__((ext_vector_type(8)));

#define WS        11
#define PAD       5
#define IMG_H     512
#define IMG_W     512
#define REG_ROWS  64           /* output rows per block (= per wave-column) */
#define REG_COLS  128          /* output cols per block (8 waves x 16 cols) */
#define TROWS     80           /* REG_ROWS + 16 halo/pad rows */
#define TCOLS     144          /* staged cols: tx0-8 .. tx0+135 (16B aligned) */
#define NSEG      36           /* TCOLS / 4 */
#define TPITCH    148          /* mult of 4; (148q)%64 distinct for q=0..15 */
#define COLOFF    3            /* LDS col of K=0 (x0-5) is 16*wv + 3 */
#define RPITCH    36           /* ring-scratch col pitch, conflict-free */
#define PSTRIDE   (16 * RPITCH) /* 576 floats between planes in scratch */
#define NWAVES    8
#define NPIX      12582912.0   /* 16*3*512*512 */
#define FXSCALE   4294967296.0 /* 2^32 fixed point -> deterministic reduction */

struct GW { float g[WS]; };

__device__ __forceinline__ float clamp01(float x) {
  return fminf(fmaxf(x, 0.f), 1.f);   // one v_max_num_f32 ... clamp
}

#define WMMA(A, B, C) __builtin_amdgcn_wmma_f32_16x16x4_f32(false, (A), false, (B), (short)0, (C), false, false)

__global__ __launch_bounds__(256) void ssim_init(unsigned long long* acc) { *acc = 0ull; }

__global__ __launch_bounds__(256) void ssim_final(const unsigned long long* acc, float* out) {
  double mean = (double)(long long)(*acc) * (1.0 / FXSCALE) / NPIX;
  *out = (float)(1.0 - mean);
}

__global__ __launch_bounds__(256) void ssim_main(const float* __restrict__ img1,
                                                 const float* __restrict__ img2,
                                                 unsigned long long* __restrict__ acc,
                                                 GW gw) {
  __shared__ __attribute__((aligned(16))) float t1[TROWS * TPITCH];
  __shared__ __attribute__((aligned(16))) float t2[TROWS * TPITCH];
  __shared__ __attribute__((aligned(16))) float scr[NWAVES * 5 * PSTRIDE];
  __shared__ float glut[48];    // g[d-15], zero outside band
  __shared__ float red[NWAVES];

  const int tid  = threadIdx.x;
  const int lane = tid & 31;
  const int wv   = tid >> 5;
  const int q    = lane & 15;   // M (A-frags) / N (B,C,D frags)
  const int h    = lane >> 4;   // lane half -> K offset 2h

  const int plane = blockIdx.z;
  const int ty0   = blockIdx.y * REG_ROWS;
  const int tx0   = blockIdx.x * REG_COLS;
  const float* p1 = img1 + (size_t)plane * (IMG_H * IMG_W);
  const float* p2 = img2 + (size_t)plane * (IMG_H * IMG_W);

  // ---- banded-Gaussian LUT: glut[d+15] = g[d] for d in [0,10], else 0 ----
  if (tid < 48) {
    float v = 0.f;
#pragma unroll
    for (int j = 0; j < WS; ++j)
      if (tid == 15 + j) v = gw.g[j];
    glut[tid] = v;
  }

  // ---- stage img tiles into LDS with CDNA5 async-to-LDS ----
  const bool interior = (ty0 >= PAD) && (ty0 + TROWS - PAD <= IMG_H) &&
                        (tx0 >= 8) && (tx0 - 8 + TCOLS <= IMG_W);
  if (interior) {
    for (int i = tid; i < TROWS * NSEG; i += 256) {
      int r = i / NSEG, s = i - r * NSEG;
      unsigned loff = (unsigned)(r * TPITCH + s * 4) * 4u;
      const float* g1 = p1 + (ty0 - PAD + r) * IMG_W + (tx0 - 8) + s * 4;
      const float* g2 = p2 + (ty0 - PAD + r) * IMG_W + (tx0 - 8) + s * 4;
      unsigned l1 = loff + (unsigned)(uintptr_t)t1;
      unsigned l2 = loff + (unsigned)(uintptr_t)t2;
      asm volatile("global_load_async_to_lds_b128 %0, %1, off" :: "v"(l1), "v"(g1) : "memory");
      asm volatile("global_load_async_to_lds_b128 %0, %1, off" :: "v"(l2), "v"(g2) : "memory");
    }
  } else {
    v4f z = {0.f, 0.f, 0.f, 0.f};
    for (int i = tid; i < TROWS * (TPITCH / 4); i += 256) {
      *(v4f*)&t1[i * 4] = z;
      *(v4f*)&t2[i * 4] = z;
    }
    __syncthreads();
    for (int i = tid; i < TROWS * NSEG; i += 256) {
      int r  = i / NSEG, s = i - r * NSEG;
      int gy = ty0 - PAD + r;
      if (gy < 0 || gy >= IMG_H) continue;
      int gx = tx0 - 8 + s * 4;
      unsigned loff = (unsigned)(r * TPITCH + s * 4) * 4u;
      unsigned l1 = loff + (unsigned)(uintptr_t)t1;
      unsigned l2 = loff + (unsigned)(uintptr_t)t2;
      const float* g1 = p1 + gy * IMG_W + gx;
      const float* g2 = p2 + gy * IMG_W + gx;
      if (gx >= 0 && gx + 4 <= IMG_W) {   // full segment
        asm volatile("global_load_async_to_lds_b128 %0, %1, off" :: "v"(l1), "v"(g1) : "memory");
        asm volatile("global_load_async_to_lds_b128 %0, %1, off" :: "v"(l2), "v"(g2) : "memory");
      } else {                            // partial edge segment
#pragma unroll
        for (int e = 0; e < 4; ++e) {
          if (gx + e >= 0 && gx + e < IMG_W) {
            unsigned le1 = l1 + 4u * e, le2 = l2 + 4u * e;
            const float* h1 = g1 + e;
            const float* h2 = g2 + e;
            asm volatile("global_load_async_to_lds_b32 %0, %1, off" :: "v"(le1), "v"(h1) : "memory");
            asm volatile("global_load_async_to_lds_b32 %0, %1, off" :: "v"(le2), "v"(h2) : "memory");
          }
        }
      }
    }
  }
  asm volatile("s_wait_asynccnt 0" ::: "memory");
  __syncthreads();

  // ---- per-lane band fragments (pass1-B == pass2-A): g[k-q], k = 4c+2h+s ----
  v2f wf[7];
#pragma unroll
  for (int c = 0; c < 7; ++c) {
    int d = 4 * c + 2 * h - q + 15;      // in [0, 42]
    v2f tmp = {glut[d], glut[d + 1]};
    wf[c] = tmp;
  }

  // ---- each wave streams a 64-row x 16-col output column ----
  float* myscr = &scr[wv * (5 * PSTRIDE)];
  float lsum = 0.f;
  const float C1 = 1.01e-4f;   // 0.01^2 + 1e-6
  const float C2 = 9.01e-4f;   // 0.03^2 + 1e-6

#pragma unroll
  for (int rt = 0; rt < 5; ++rt) {
    // ===== pass 1, row-tile rt: T rows 16rt..16rt+15 -> ring slot rt&1 =====
    {
      const float* tb1 = &t1[(rt * 16 + q) * TPITCH + 16 * wv + COLOFF + 2 * h];
      const float* tb2 = &t2[(rt * 16 + q) * TPITCH + 16 * wv + COLOFF + 2 * h];
      float x1[14], x2[14];
#pragma unroll
      for (int c = 0; c < 7; ++c) {      // fuses to ds_load_2addr_b32
        x1[2 * c]     = tb1[4 * c];
        x1[2 * c + 1] = tb1[4 * c + 1];
        x2[2 * c]     = tb2[4 * c];
        x2[2 * c + 1] = tb2[4 * c + 1];
      }
      v8f tac[5];
#pragma unroll
      for (int p = 0; p < 5; ++p) {
        v8f zz = {0.f, 0.f, 0.f, 0.f, 0.f, 0.f, 0.f, 0.f};
        tac[p] = zz;
      }
#pragma unroll
      for (int c = 0; c < 7; ++c) {
        float a0 = clamp01(x1[2 * c]), a1 = clamp01(x1[2 * c + 1]);
        float b0 = clamp01(x2[2 * c]), b1 = clamp01(x2[2 * c + 1]);
        v2f f0 = {a0, a1};
        v2f f1 = {b0, b1};
        v2f f2 = {a0 * a0, a1 * a1};
        v2f f3 = {b0 * b0, b1 * b1};
        v2f f4 = {a0 * b0, a1 * b1};
        tac[0] = WMMA(f0, wf[c], tac[0]);
        tac[1] = WMMA(f1, wf[c], tac[1]);
        tac[2] = WMMA(f2, wf[c], tac[2]);
        tac[3] = WMMA(f3, wf[c], tac[3]);
        tac[4] = WMMA(f4, wf[c], tac[4]);
      }
      // D -> ring scratch through v4f-typed pointer: ds_store_b128 with
      // 16-bit imm offset absorbing p*PSTRIDE -> one base, zero v_adds
      v4f* spv = (v4f*)&myscr[q * RPITCH + 16 * (rt & 1) + 8 * h];  // 16B aligned
#pragma unroll
      for (int p = 0; p < 5; ++p) {
        spv[p * (PSTRIDE / 4)]     = __builtin_shufflevector(tac[p], tac[p], 0, 1, 2, 3);
        spv[p * (PSTRIDE / 4) + 1] = __builtin_shufflevector(tac[p], tac[p], 4, 5, 6, 7);
      }
    }

    // ===== pass 2, output tile j = rt-1: reads ring rows 16j..16j+27 =====
    if (rt >= 1) {
      const int j = rt - 1;
      // v2f-typed scratch reads: ds_load_b64, 16-bit imm offset, one base
      const v2f* sbv = (const v2f*)&myscr[q * RPITCH + 2 * h];      // 8B aligned
      v8f r5[5];
#pragma unroll
      for (int p = 0; p < 5; ++p) {
        v2f bv[7];
#pragma unroll
        for (int c = 0; c < 7; ++c) {
          int lo = (16 * (j & 1) + 4 * c) & 31;  // even; v2f index = lo/2
          bv[c] = sbv[p * (PSTRIDE / 2) + lo / 2];
        }
        v8f a = {0.f, 0.f, 0.f, 0.f, 0.f, 0.f, 0.f, 0.f};
#pragma unroll
        for (int c = 0; c < 7; ++c)
          a = WMMA(wf[c], bv[c], a);
        r5[p] = a;
      }
      // elementwise SSIM for this 16x16 tile (hw rcp: den >= C1*C2 > 0)
#pragma unroll
      for (int v = 0; v < 8; ++v) {
        float mu1 = r5[0][v], mu2 = r5[1][v];
        float e11 = r5[2][v], e22 = r5[3][v], e12 = r5[4][v];
        float mu1s = mu1 * mu1, mu2s = mu2 * mu2, m12 = mu1 * mu2;
        float s1  = fminf(fmaxf(e11 - mu1s, 1e-6f), 1e6f);
        float s2  = fminf(fmaxf(e22 - mu2s, 1e-6f), 1e6f);
        float s12 = e12 - m12;
        float num = (2.f * m12 + C1) * (2.f * s12 + C2);
        float den = (mu1s + mu2s + C1) * (s1 + s2 + C2);
        float ssim = num * __builtin_amdgcn_rcpf(den);
        if (ssim != ssim) ssim = 0.f;
        lsum += ssim;
      }
    }
  }

  // ---- deterministic fixed-point reduction ----
  for (int off = 16; off > 0; off >>= 1)
    lsum += __shfl_xor(lsum, off, 32);
  if (lane == 0) red[wv] = lsum;
  __syncthreads();
  if (tid == 0) {
    float bs = 0.f;
    for (int i = 0; i < NWAVES; ++i) bs += red[i];
    long long fx = __double2ll_rn((double)bs * FXSCALE);
    atomicAdd(acc, (unsigned long long)fx);
  }
}

extern "C" void kernel_launch(void* const* d_in, const int* in_sizes, int n_in,
                              void* d_out, int out_size, void* d_ws, size_t ws_size,
                              hipStream_t stream) {
  (void)in_sizes; (void)n_in; (void)out_size; (void)ws_size;
  const float* img1 = (const float*)d_in[0];
  const float* img2 = (const float*)d_in[1];
  float* out = (float*)d_out;
  unsigned long long* acc = (unsigned long long*)d_ws;

  GW gw;
  double gd[WS], s = 0.0;
  for (int i = 0; i < WS; ++i) {
    double d = (double)(i - WS / 2);
    gd[i] = exp(-(d * d) / (2.0 * 1.5 * 1.5));
    s += gd[i];
  }
  for (int i = 0; i < WS; ++i) gw.g[i] = (float)(gd[i] / s);

  hipLaunchKernelGGL(ssim_init, dim3(1), dim3(1), 0, stream, acc);
  dim3 grid(IMG_W / REG_COLS, IMG_H / REG_ROWS, 48);  // (4, 8, 48)
  hipLaunchKernelGGL(ssim_main, grid, dim3(256), 0, stream, img1, img2, acc, gw);
  hipLaunchKernelGGL(ssim_final, dim3(1), dim3(1), 0, stream, acc, out);
}